// ROIPooling_1932735283567
// MI455X (gfx1250) — compile-verified
//
#include <hip/hip_runtime.h>

// Adaptive max pool (512,256,28,28) -> (512,256,7,7).
// With in=28, out=7 the adaptive bins are exact non-overlapping 4x4 windows:
//   start = floor(i*28/7) = 4i, end = ceil((i+1)*28/7) = 4(i+1), no masking.
// Pure streaming kernel: ~437 MB total traffic -> ~18.7 us floor at 23.3 TB/s.
// One thread per output element; four 16B (float4) non-temporal loads
// (all 16B-aligned), 15 max ops, one non-temporal store, plus a speculative
// global_prefetch_b8 lookahead into GL2.
// Index math is UNSIGNED so /49 and /7 lower to umulhi+shift without the
// signed-division sign-fixup chains seen in the previous round's disasm.

typedef float v4f __attribute__((ext_vector_type(4)));

__device__ __forceinline__ v4f vmax4(v4f a, v4f b) {
  v4f r;
  r.x = fmaxf(a.x, b.x);
  r.y = fmaxf(a.y, b.y);
  r.z = fmaxf(a.z, b.z);
  r.w = fmaxf(a.w, b.w);
  return r;
}

__global__ __launch_bounds__(256) void adaptive_max_pool_28to7_kernel(
    const float* __restrict__ in, float* __restrict__ out, unsigned total) {
  unsigned tid = blockIdx.x * blockDim.x + threadIdx.x;
  if (tid >= total) return;

  // tid -> (plane, oh, ow); plane = n*C + c, 49 outputs per plane.
  unsigned plane = tid / 49u;
  unsigned rem   = tid - plane * 49u;
  unsigned oh    = rem / 7u;
  unsigned ow    = rem - oh * 7u;

  // Window top-left: row 4*oh, col 4*ow. Row stride 28 floats (112 B),
  // plane stride 784 floats (3136 B). All float4 loads below are 16B-aligned.
  const float* base = in + (size_t)plane * 784u + (oh * 112u + ow * 4u);

  // Speculative lookahead (~75 KB ahead of the stream) into GL2; OOB
  // prefetches are silently dropped per the ISA (speculative TH).
  __builtin_prefetch(base + 784 * 24, /*rw=*/0, /*locality=*/0);

  // Four rows of the 4x4 window as float4, non-temporal (single-use stream;
  // don't rinse regular-temporal lines through L2).
  v4f r0 = __builtin_nontemporal_load((const v4f*)(base));
  v4f r1 = __builtin_nontemporal_load((const v4f*)(base + 28));
  v4f r2 = __builtin_nontemporal_load((const v4f*)(base + 56));
  v4f r3 = __builtin_nontemporal_load((const v4f*)(base + 84));

  v4f m = vmax4(vmax4(r0, r1), vmax4(r2, r3));
  float res = fmaxf(fmaxf(m.x, m.y), fmaxf(m.z, m.w));

  __builtin_nontemporal_store(res, out + tid);
}

extern "C" void kernel_launch(void* const* d_in, const int* in_sizes, int n_in,
                              void* d_out, int out_size, void* d_ws, size_t ws_size,
                              hipStream_t stream) {
  (void)in_sizes; (void)n_in; (void)d_ws; (void)ws_size;
  const float* in = (const float*)d_in[0];
  float* out = (float*)d_out;

  unsigned total = (unsigned)out_size;  // N*C*7*7 output elements
  const int block = 256;                // 8 wave32 waves per block
  unsigned grid = (total + block - 1) / block;
  adaptive_max_pool_28to7_kernel<<<grid, block, 0, stream>>>(in, out, total);
}